// StickBreakingSegmentation_17179869184694
// MI455X (gfx1250) — compile-verified
//
#include <hip/hip_runtime.h>
#include <cstddef>

typedef __attribute__((ext_vector_type(2))) float v2f;
typedef __attribute__((ext_vector_type(8))) float v8f;

#define B_      32
#define CIN     64
#define COUT    8
#define H_      256
#define W_      256
#define HW      65536
#define KSTEPS  11

// ---------------------------------------------------------------------------
// Kernel 1: feat = gate * (conv_w @ x + b) + coord_grid, via f32 WMMA.
// One wave computes a 16(ch-pad) x 16(pixel) tile with 16 chained
// v_wmma_f32_16x16x4_f32 ops (K=64). Channel-major output -> coalesced.
// ---------------------------------------------------------------------------
__global__ __launch_bounds__(256) void feat_kernel(
    const float* __restrict__ x,       // [B, CIN, HW]
    const float* __restrict__ w,       // [COUT, CIN]
    const float* __restrict__ bias,    // [COUT]
    const float* __restrict__ gate_p,  // scalar
    float* __restrict__ feat)          // [B, COUT, HW]
{
    const int lane     = threadIdx.x & 31;            // wave32
    const int tile     = blockIdx.x * 8 + (threadIdx.x >> 5);
    const int b        = tile >> 12;                  // tile / (HW/16)
    const int pix_base = (tile & 4095) << 4;

    const int half = (lane & 16) ? 2 : 0;             // K split per ISA layout
    const int mrow = lane & 15;                       // A row / B column

    // Preload all 16 A-fragments of conv_w (rows 8..15 zero-padded).
    v2f a[16];
#pragma unroll
    for (int kk = 0; kk < 16; ++kk) {
        const int kb = kk * 4 + half;
        if (mrow < COUT) {
            a[kk].x = w[mrow * CIN + kb];
            a[kk].y = w[mrow * CIN + kb + 1];
        } else {
            a[kk].x = 0.0f;
            a[kk].y = 0.0f;
        }
    }

    const float* xb = x + (size_t)b * CIN * HW;
    v8f c = {};
#pragma unroll
    for (int kk = 0; kk < 16; ++kk) {
        const int row = kk * 4 + half;
        const float* bp = xb + (size_t)row * HW + pix_base + mrow;
        v2f bf;
        bf.x = bp[0];      // K = row
        bf.y = bp[HW];     // K = row + 1
        // (neg_a, A, neg_b, B, c_mod, C, reuse_a, reuse_b)
        c = __builtin_amdgcn_wmma_f32_16x16x4_f32(
                false, a[kk], false, bf, (short)0, c, false, false);
    }

    const float gate = gate_p[0];
    if (lane < 16) {                  // D: VGPR r = channel r, lane = pixel
        const int p = pix_base + lane;
        const int i = p >> 8;         // row (W_ == 256)
        const int j = p & 255;        // col
        const float gx = -1.0f + (2.0f / 255.0f) * (float)i;
        const float gy = -1.0f + (2.0f / 255.0f) * (float)j;
#pragma unroll
        for (int r = 0; r < COUT; ++r) {
            float v = gate * (c[r] + bias[r]);
            if (r == COUT - 2) v += gx;
            if (r == COUT - 1) v += gy;
            feat[(size_t)(b * COUT + r) * HW + p] = v;
        }
    }
}

// ---------------------------------------------------------------------------
// Kernel 2: per-batch argmax of rand*scope (first-index tie-break, matching
// jnp.argmax), then gather the 8-channel seed. One 1024-thread block / batch.
// ---------------------------------------------------------------------------
__global__ __launch_bounds__(1024) void argmax_kernel(
    const float* __restrict__ rand_pixel,  // [B, HW]
    const float* __restrict__ scope_buf,   // [B, HW]
    const float* __restrict__ feat,        // [B, COUT, HW]
    float* __restrict__ seed,              // [B, COUT]
    int step)
{
    __shared__ float sv[1024];
    __shared__ int   si[1024];
    const int b = blockIdx.x;
    const int t = threadIdx.x;
    const float* rb = rand_pixel + (size_t)b * HW;
    const float* sb = scope_buf  + (size_t)b * HW;

    float best = -1.0f;
    int   bidx = 0;
    for (int p = t; p < HW; p += 1024) {
        const float v = rb[p] * (step == 0 ? 1.0f : sb[p]);
        if (v > best) { best = v; bidx = p; }   // ascending p => first max kept
    }
    sv[t] = best;
    si[t] = bidx;
    __syncthreads();

    for (int s = 512; s > 0; s >>= 1) {
        if (t < s) {
            const float v2 = sv[t + s];
            const int   i2 = si[t + s];
            if (v2 > sv[t] || (v2 == sv[t] && i2 < si[t])) {
                sv[t] = v2;
                si[t] = i2;
            }
        }
        __syncthreads();
    }

    if (t < COUT) {
        const int idx = si[0];
        seed[b * COUT + t] = feat[(size_t)(b * COUT + t) * HW + idx];
    }
}

// ---------------------------------------------------------------------------
// Kernel 3: stick-breaking step in multiplicative form.
//   alpha  = clip(exp(-|feat - seed|^2 / sigma), 0.01, 0.99)
//   masks[b,k] = scope*alpha ; scopes[b,k] = scope ; scope *= (1 - alpha)
// Last step also writes slice K-1 (= final scope) of both outputs.
// ---------------------------------------------------------------------------
__global__ __launch_bounds__(256) void update_kernel(
    const float* __restrict__ feat,        // [B, COUT, HW]
    const float* __restrict__ seed,        // [B, COUT]
    const float* __restrict__ log_sigma_p, // scalar
    float* __restrict__ scope_buf,         // [B, HW]
    float* __restrict__ masks,             // [B, KSTEPS, HW]
    float* __restrict__ scopes,            // [B, KSTEPS, HW]
    int step)
{
    const int gid = blockIdx.x * 256 + threadIdx.x;
    const int b   = gid >> 16;        // / HW
    const int p   = gid & (HW - 1);

    const float inv_sigma = expf(-log_sigma_p[0]);   // 1/sigma = exp(-log_sigma)
    const float sc = (step == 0) ? 1.0f : scope_buf[gid];

    const float* fb = feat + (size_t)b * COUT * HW + p;
    const float* sd = seed + b * COUT;
    float sq = 0.0f;
#pragma unroll
    for (int c = 0; c < COUT; ++c) {
        const float d = fb[(size_t)c * HW] - sd[c];
        sq += d * d;
    }

    float alpha = expf(-sq * inv_sigma);
    alpha = fminf(fmaxf(alpha, 0.01f), 0.99f);       // straight-through forward

    const size_t ob = ((size_t)b * KSTEPS + step) * HW + p;
    masks[ob]  = sc * alpha;
    scopes[ob] = sc;

    const float ns = sc * (1.0f - alpha);
    scope_buf[gid] = ns;

    if (step == KSTEPS - 2) {                        // write final K-1 slice
        const size_t of = ((size_t)b * KSTEPS + (KSTEPS - 1)) * HW + p;
        masks[of]  = ns;
        scopes[of] = ns;
    }
}

// ---------------------------------------------------------------------------
extern "C" void kernel_launch(void* const* d_in, const int* in_sizes, int n_in,
                              void* d_out, int out_size, void* d_ws, size_t ws_size,
                              hipStream_t stream) {
    const float* x          = (const float*)d_in[0];  // [B, CIN, H, W]
    const float* rand_pixel = (const float*)d_in[1];  // [B, 1, H, W]
    const float* conv_w     = (const float*)d_in[2];  // [COUT, CIN]
    const float* conv_b     = (const float*)d_in[3];  // [COUT]
    const float* gate       = (const float*)d_in[4];  // scalar
    const float* log_sigma  = (const float*)d_in[5];  // scalar

    float* masks  = (float*)d_out;                            // [B, K, H, W]
    float* scopes = masks + (size_t)B_ * KSTEPS * HW;         // [B, K, H, W]

    float* feat      = (float*)d_ws;                          // 16,777,216 f
    float* scope_buf = feat + (size_t)B_ * COUT * HW;         //  2,097,152 f
    float* seed      = scope_buf + (size_t)B_ * HW;           //        256 f

    // Phase 1: WMMA 1x1 conv -> feat.  B*HW/16 = 131072 wave-tiles, 8/block.
    feat_kernel<<<16384, 256, 0, stream>>>(x, conv_w, conv_b, gate, feat);

    // Phase 2: 10 serial stick-breaking steps (global argmax dependency).
    for (int k = 0; k < KSTEPS - 1; ++k) {
        argmax_kernel<<<B_, 1024, 0, stream>>>(rand_pixel, scope_buf, feat, seed, k);
        update_kernel<<<(B_ * HW) / 256, 256, 0, stream>>>(
            feat, seed, log_sigma, scope_buf, masks, scopes, k);
    }
}